// SparseNGCNLayer_25288767439532
// MI455X (gfx1250) — compile-verified
//
#include <hip/hip_runtime.h>
#include <hip/hip_bf16.h>

#define N_NODES   100000
#define IN_CH     2048
#define OUT_CH    128
#define NNZ_FEAT  2000000
#define NNZ_ADJ   1600000

typedef __attribute__((ext_vector_type(2))) float v2f;
typedef __attribute__((ext_vector_type(8))) float v8f;

// ---------------------------------------------------------------------------
// Zero a float buffer (own kernel: guaranteed graph-capture safe).
// ---------------------------------------------------------------------------
__global__ void zero_f32_kernel(float4* __restrict__ p, int n4) {
    int i = blockIdx.x * blockDim.x + threadIdx.x;
    const int stride = gridDim.x * blockDim.x;
    for (; i < n4; i += stride) p[i] = make_float4(0.f, 0.f, 0.f, 0.f);
}

// ---------------------------------------------------------------------------
// Generic COO SpMM scatter: dst[rows[k], :] += vals[k] * dense[cols[k], :]
// One wave per nnz: 32 lanes x float4 = 128 consecutive f32.
// Gather is a coalesced 512B row read (L2-resident); scatter uses hardware
// f32 atomics (global_atomic_add_f32, no-return -> STOREcnt).
// ---------------------------------------------------------------------------
__device__ __forceinline__ void atomic_add_f32(float* p, float v) {
    unsafeAtomicAdd(p, v);   // force hardware global_atomic_add_f32
}

__global__ void spmm_scatter_kernel(const int*   __restrict__ rows,
                                    const int*   __restrict__ cols,
                                    const float* __restrict__ vals,
                                    const float* __restrict__ dense,  // [*, 128]
                                    float*       __restrict__ dst,    // [N, 128]
                                    int nnz) {
    const int lane          = threadIdx.x & 31;
    const int waveInBlock   = threadIdx.x >> 5;
    const int wavesPerBlock = blockDim.x >> 5;
    const int wave          = blockIdx.x * wavesPerBlock + waveInBlock;
    const int nWaves        = gridDim.x * wavesPerBlock;

    for (int k = wave; k < nnz; k += nWaves) {
        // Stream the next iteration's COO entries toward the caches.
        __builtin_prefetch(rows + k + nWaves, 0, 0);
        __builtin_prefetch(cols + k + nWaves, 0, 0);
        __builtin_prefetch(vals + k + nWaves, 0, 0);

        const int   r = rows[k];            // broadcast load (all lanes same addr)
        const int   c = cols[k];
        const float v = vals[k];

        const float4 w =
            ((const float4*)(dense + (size_t)c * OUT_CH))[lane];   // 512B coalesced
        float* out = dst + (size_t)r * OUT_CH + lane * 4;
        atomic_add_f32(out + 0, v * w.x);
        atomic_add_f32(out + 1, v * w.y);
        atomic_add_f32(out + 2, v * w.z);
        atomic_add_f32(out + 3, v * w.w);
    }
}

// ---------------------------------------------------------------------------
// Bias + ReLU epilogue via WMMA: h_tile(16x16) = relu( 1*bias + h_tile ).
// D = A x B + C with A = e0 column of ones (A[m][0]=1), B[0][n] = bias[n].
// One wave per 16x16 tile. N_NODES=100000 is divisible by 16; 128/16=8 col
// tiles. Full waves, no divergence -> EXEC all-ones as WMMA requires.
// ---------------------------------------------------------------------------
__global__ void bias_relu_wmma_kernel(float* __restrict__ h,        // [N, 128]
                                      const float* __restrict__ bias) { // [128]
    const int lane          = threadIdx.x & 31;
    const int waveInBlock   = threadIdx.x >> 5;
    const int wavesPerBlock = blockDim.x >> 5;
    const int tile          = blockIdx.x * wavesPerBlock + waveInBlock;

    const int nTiles = (N_NODES / 16) * (OUT_CH / 16);
    if (tile >= nTiles) return;

    const int rowTile = tile >> 3;        // 8 column tiles per row block
    const int colTile = tile & 7;
    const int r0 = rowTile * 16;
    const int c0 = colTile * 16;

    const int nsub  = lane & 15;          // N index within tile
    const int mbase = (lane >> 4) * 8;    // lanes 0-15: M=j ; lanes 16-31: M=8+j

    // Load C (16x16 f32 accumulator layout: VGPR j <-> M = mbase + j, N = nsub)
    v8f cacc;
#pragma unroll
    for (int j = 0; j < 8; ++j)
        cacc[j] = h[(size_t)(r0 + mbase + j) * OUT_CH + c0 + nsub];

    // A (16x4): lanes 0-15 hold (K0,K1) for M=lane; lanes 16-31 hold (K2,K3).
    v2f a;
    a.x = (lane < 16) ? 1.0f : 0.0f;      // A[m][0] = 1
    a.y = 0.0f;                           // A[m][1] = A[m][2] = A[m][3] = 0

    // B (4x16): VGPR0 lanes 0-15 = row K0 (bias), lanes 16-31 = row K2 (zero).
    v2f b;
    b.x = (lane < 16) ? bias[c0 + nsub] : 0.0f;
    b.y = 0.0f;                           // rows K1 / K3 = 0

    v8f d = __builtin_amdgcn_wmma_f32_16x16x4_f32(
        /*neg_a=*/false, a, /*neg_b=*/false, b,
        /*c_mod=*/(short)0, cacc, /*reuse_a=*/false, /*reuse_b=*/false);

#pragma unroll
    for (int j = 0; j < 8; ++j)
        h[(size_t)(r0 + mbase + j) * OUT_CH + c0 + nsub] = fmaxf(d[j], 0.0f);
}

// ---------------------------------------------------------------------------
// Host-side launch. Ping-pong h between d_ws and d_out so the final
// propagation step lands in d_out. Destinations are zeroed before each
// scatter accumulation (harness poisons buffers; replays must be
// deterministic).
// ---------------------------------------------------------------------------
extern "C" void kernel_launch(void* const* d_in, const int* in_sizes, int n_in,
                              void* d_out, int out_size, void* d_ws, size_t ws_size,
                              hipStream_t stream) {
    const int*   feat_rows = (const int*)  d_in[0];
    const int*   feat_cols = (const int*)  d_in[1];
    const float* feat_vals = (const float*)d_in[2];
    const int*   adj_rows  = (const int*)  d_in[3];
    const int*   adj_cols  = (const int*)  d_in[4];
    const float* adj_vals  = (const float*)d_in[5];
    const float* weight    = (const float*)d_in[6];  // [2048, 128]
    const float* bias      = (const float*)d_in[7];  // [1, 128]

    float* hA = (float*)d_ws;    // scratch h buffer [N, 128]
    float* hB = (float*)d_out;   // final output     [N, 128]

    const int n4 = (N_NODES * OUT_CH) / 4;   // float4 count
    const dim3 zgrid(2048), zblk(256);
    const dim3 sgrid(4096), sblk(256);       // 32768 waves, grid-stride over nnz
    const int  nTiles = (N_NODES / 16) * (OUT_CH / 16);   // 50000
    const dim3 bgrid((nTiles + 7) / 8), bblk(256);        // 8 waves/block

    // h0 = relu( SpMM(feat, W) + bias )  -> hA
    zero_f32_kernel<<<zgrid, zblk, 0, stream>>>((float4*)hA, n4);
    spmm_scatter_kernel<<<sgrid, sblk, 0, stream>>>(
        feat_rows, feat_cols, feat_vals, weight, hA, NNZ_FEAT);
    bias_relu_wmma_kernel<<<bgrid, bblk, 0, stream>>>(hA, bias);

    // h1 = SpMM(adj, h0) -> hB
    zero_f32_kernel<<<zgrid, zblk, 0, stream>>>((float4*)hB, n4);
    spmm_scatter_kernel<<<sgrid, sblk, 0, stream>>>(
        adj_rows, adj_cols, adj_vals, hA, hB, NNZ_ADJ);

    // h2 = SpMM(adj, h1) -> hA
    zero_f32_kernel<<<zgrid, zblk, 0, stream>>>((float4*)hA, n4);
    spmm_scatter_kernel<<<sgrid, sblk, 0, stream>>>(
        adj_rows, adj_cols, adj_vals, hB, hA, NNZ_ADJ);

    // h3 = SpMM(adj, h2) -> hB == d_out
    zero_f32_kernel<<<zgrid, zblk, 0, stream>>>((float4*)hB, n4);
    spmm_scatter_kernel<<<sgrid, sblk, 0, stream>>>(
        adj_rows, adj_cols, adj_vals, hA, hB, NNZ_ADJ);
}